// MultiScaleFlashAttn_80788334838526
// MI455X (gfx1250) — compile-verified
//
#include <hip/hip_runtime.h>
#include <math.h>

typedef float v2f __attribute__((ext_vector_type(2)));
typedef float v8f __attribute__((ext_vector_type(8)));
typedef unsigned int u32x4 __attribute__((ext_vector_type(4)));
typedef int i32x4 __attribute__((ext_vector_type(4)));
typedef int i32x8 __attribute__((ext_vector_type(8)));

// Problem constants (structural, from the reference)
#define NB     2
#define QN     12240
#define HWTOT  12240
#define ROWS   (NB * QN)        // 24480, multiple of 16
#define DIMS   256
#define HEADS  8
#define CH     32
#define LEV    4
#define PTS    4

#define BN         64           // column-tile width per block (and per wave)
#define KDIM       256          // K is 256 in every GEMM here
#define LDSSTRIDE  72           // 64 data + 8 pad floats; 2*72 % 64 == 16 -> the two
                                // lane-halves hit disjoint LDS bank ranges

// -----------------------------------------------------------------------------
// Fp32 WMMA GEMM:  C[M x N] = A[M x K] * B[K x N] + bias[N],  K == 256.
// Block = 8 waves. The block's 64-wide B panel (K x 64 floats) is DMA'd into
// LDS once by the Tensor Data Mover (tensor_load_to_lds, TENSORcnt), with
// hardware padding of 8 DWORDs per 64-DWORD row for bank-conflict-free reads.
// Each wave then computes a 16x64 output strip with V_WMMA_F32_16X16X4_F32,
// A fragments from global (L2-resident), B fragments from LDS.
// -----------------------------------------------------------------------------
__global__ __launch_bounds__(256) void gemm_bias_wmma_f32(
    const float* __restrict__ A, const float* __restrict__ B,
    const float* __restrict__ bias, float* __restrict__ C,
    int M, int N)
{
    __shared__ float Bs[KDIM * LDSSTRIDE];   // 73,728 B of the 320 KB WGP LDS

    const int ctiles = N >> 6;               // number of 64-wide column tiles
    const int tcb    = blockIdx.x % ctiles;  // this block's column tile
    const int tmb    = blockIdx.x / ctiles;  // this block's group of 8 M-tiles
    const int wid    = threadIdx.x >> 5;
    const int tm     = tmb * 8 + wid;        // this wave's 16-row M tile

    // ---- TDM: async-DMA the B panel [K x 64] into LDS (wave 0 only) ----
    if (wid == 0) {
        const unsigned long long gaddr =
            (unsigned long long)(uintptr_t)B + (unsigned long long)tcb * BN * 4ull;
        const unsigned int ldsoff = (unsigned int)(uintptr_t)(void*)&Bs[0];

        u32x4 g0;
        g0.x = 1u;                                   // count=1 (valid descriptor)
        g0.y = ldsoff;                               // lds_addr (bytes)
        g0.z = (unsigned int)gaddr;                  // global_addr[31:0]
        g0.w = (unsigned int)(gaddr >> 32) | (2u << 30);  // global_addr[56:32] | type=2

        i32x8 g1;
        g1[0] = (2 << 16)                            // data_size = 4 bytes
              | (1 << 20)                            // pad_enable
              | (5 << 22)                            // pad_interval: 64 DWORDs
              | (7 << 25);                           // pad_amount: 8 DWORDs
        g1[1] = (N & 0xFFFF) << 16;                  // tensor_dim0[15:0]
        g1[2] = ((unsigned)N >> 16) | ((KDIM & 0xFFFF) << 16);   // dim0 hi | dim1 lo
        g1[3] = (KDIM >> 16) | (BN << 16);           // dim1 hi | tile_dim0 = 64
        g1[4] = KDIM;                                // tile_dim1 = K, tile_dim2 = 0
        g1[5] = N;                                   // tensor_dim0_stride[31:0]
        g1[6] = 0;                                   // stride hi | dim1_stride lo
        g1[7] = 0;

        i32x4 z4 = {0, 0, 0, 0};
#if __has_include(<hip/amd_detail/amd_gfx1250_TDM.h>)
        i32x8 z8 = {0, 0, 0, 0, 0, 0, 0, 0};
        __builtin_amdgcn_tensor_load_to_lds(g0, g1, z4, z4, z8, 0);
#else
        __builtin_amdgcn_tensor_load_to_lds(g0, g1, z4, z4, 0);
#endif
        __builtin_amdgcn_s_wait_tensorcnt(0);
    }
    __syncthreads();                                 // publish LDS to all 8 waves

    if (tm * 16 < M) {                               // wave-uniform (EXEC all-1 for WMMA)
        const int lane = threadIdx.x & 31;
        const int half = lane >> 4;                  // 0: K=0,1   1: K=2,3
        const int lid  = lane & 15;

        const float* Arow = A + (size_t)(tm * 16 + lid) * (size_t)KDIM;

        v8f acc[4];
#pragma unroll
        for (int s = 0; s < 4; ++s) acc[s] = (v8f){0.f,0.f,0.f,0.f,0.f,0.f,0.f,0.f};

        for (int k = 0; k < KDIM; k += 4) {
            const int ka = k + half * 2;
            v2f a;
            a.x = Arow[ka];
            a.y = Arow[ka + 1];
#pragma unroll
            for (int s = 0; s < 4; ++s) {
                const int bcl = s * 16 + lid;        // column within the 64-wide panel
                v2f b;
                b.x = Bs[ka * LDSSTRIDE + bcl];
                b.y = Bs[(ka + 1) * LDSSTRIDE + bcl];
                acc[s] = __builtin_amdgcn_wmma_f32_16x16x4_f32(
                    false, a, false, b, (short)0, acc[s], false, false);
            }
        }

        // C/D layout: VGPR r -> M = r (lanes 0-15) / M = 8+r (lanes 16-31), N = lid
        const int r0 = tm * 16 + half * 8;
#pragma unroll
        for (int s = 0; s < 4; ++s) {
            const int cc = tcb * BN + s * 16 + lid;
            const float bv = bias[cc];
#pragma unroll
            for (int r = 0; r < 8; ++r)
                C[(size_t)(r0 + r) * N + cc] = acc[s][r] + bv;
        }
    }
}

// -----------------------------------------------------------------------------
// Deformable sampling: one wave per (n, q, head); lane = channel (CH == 32).
// Softmax over the 16 (level,point) logits, then 4-corner bilinear gathers of
// the projected value tensor, weighted-summed into out_pre[n,q,h*32+lane].
// Location math is wave-uniform; gathers are 32-channel coalesced dword loads.
// -----------------------------------------------------------------------------
__global__ __launch_bounds__(256) void msda_sample_kernel(
    const float* __restrict__ vproj,   // [NB, HWTOT, 256] head-major channels
    const float* __restrict__ offw,    // [NB, QN, 256] = (H,L,P,2)
    const float* __restrict__ attnw,   // [NB, QN, 128] = (H,L*P) logits
    const float* __restrict__ pref,    // [NB, QN, LEV, 2]
    float* __restrict__ outp)          // [NB, QN, 256]
{
    const int   Hs[LEV]  = {96, 48, 24, 12};
    const int   Wsz[LEV] = {96, 48, 24, 12};
    const int   st[LEV]  = {0, 9216, 11520, 12096};

    const int wid = threadIdx.x >> 5;
    const int gw  = blockIdx.x * 8 + wid;
    if (gw >= NB * QN * HEADS) return;
    const int h = gw & 7;
    const int q = (gw >> 3) % QN;
    const int n = gw / (8 * QN);
    const int lane = threadIdx.x & 31;

    // ---- softmax over 16 logits for this (n,q,h) ----
    const float* lg = attnw + (size_t)(n * QN + q) * (HEADS * LEV * PTS) + h * (LEV * PTS);
    float w[LEV * PTS];
    float mx = -3.0e38f;
#pragma unroll
    for (int i = 0; i < LEV * PTS; ++i) { w[i] = lg[i]; mx = fmaxf(mx, w[i]); }
    float se = 0.f;
#pragma unroll
    for (int i = 0; i < LEV * PTS; ++i) { w[i] = __expf(w[i] - mx); se += w[i]; }
    const float inv = 1.f / se;

    const float* ofs = offw + (size_t)(n * QN + q) * DIMS + h * (LEV * PTS * 2);
    const float* pr  = pref + (size_t)(n * QN + q) * (LEV * 2);
    const float* vb  = vproj + (size_t)n * HWTOT * DIMS + h * CH + lane;

    float acc = 0.f;
#pragma unroll
    for (int l = 0; l < LEV; ++l) {
        const int   Hl = Hs[l], Wl = Wsz[l];
        const float fW = (float)Wl, fH = (float)Hl;
        const float px = pr[l * 2 + 0];
        const float py = pr[l * 2 + 1];
        const float* vlev = vb + (size_t)st[l] * DIMS;
#pragma unroll
        for (int pt = 0; pt < PTS; ++pt) {
            const float ox = ofs[l * (PTS * 2) + pt * 2 + 0];
            const float oy = ofs[l * (PTS * 2) + pt * 2 + 1];
            const float lx = px + ox / fW;            // off normalized by (W, H)
            const float ly = py + oy / fH;
            const float x  = lx * fW - 0.5f;          // align_corners=False
            const float y  = ly * fH - 0.5f;
            const float x0f = floorf(x), y0f = floorf(y);
            const float dx = x - x0f,  dy = y - y0f;
            const int   x0 = (int)x0f, y0 = (int)y0f;
            const int   x1 = x0 + 1,   y1 = y0 + 1;

            float s = 0.f;
            if (x0 >= 0 && x0 < Wl && y0 >= 0 && y0 < Hl)
                s += (1.f - dx) * (1.f - dy) * vlev[(size_t)(y0 * Wl + x0) * DIMS];
            if (x1 >= 0 && x1 < Wl && y0 >= 0 && y0 < Hl)
                s += dx * (1.f - dy) * vlev[(size_t)(y0 * Wl + x1) * DIMS];
            if (x0 >= 0 && x0 < Wl && y1 >= 0 && y1 < Hl)
                s += (1.f - dx) * dy * vlev[(size_t)(y1 * Wl + x0) * DIMS];
            if (x1 >= 0 && x1 < Wl && y1 >= 0 && y1 < Hl)
                s += dx * dy * vlev[(size_t)(y1 * Wl + x1) * DIMS];

            acc += w[l * PTS + pt] * inv * s;
        }
    }
    outp[(size_t)(n * QN + q) * DIMS + h * CH + lane] = acc;
}

// -----------------------------------------------------------------------------
// Launch: 3 WMMA GEMMs -> sampling -> final WMMA GEMM
// -----------------------------------------------------------------------------
extern "C" void kernel_launch(void* const* d_in, const int* in_sizes, int n_in,
                              void* d_out, int out_size, void* d_ws, size_t ws_size,
                              hipStream_t stream) {
    (void)in_sizes; (void)n_in; (void)out_size; (void)ws_size;

    const float* q      = (const float*)d_in[0];
    const float* p      = (const float*)d_in[1];
    const float* v      = (const float*)d_in[2];
    const float* W_off  = (const float*)d_in[3];
    const float* b_off  = (const float*)d_in[4];
    const float* W_attn = (const float*)d_in[5];
    const float* b_attn = (const float*)d_in[6];
    const float* W_in   = (const float*)d_in[7];
    const float* b_in   = (const float*)d_in[8];
    const float* W_out  = (const float*)d_in[9];
    const float* b_out  = (const float*)d_in[10];
    float* out = (float*)d_out;

    // workspace layout (floats): vproj | off | attn | out_pre  (~88 MB total)
    float* ws    = (float*)d_ws;
    float* vproj = ws;
    float* offw  = vproj + (size_t)ROWS * DIMS;
    float* attnw = offw  + (size_t)ROWS * DIMS;
    float* prew  = attnw + (size_t)ROWS * (HEADS * LEV * PTS);

    const int THREADS = 256;                     // 8 waves per block (wave32)
    auto gemm_grid = [](int M, int N) {
        int ctiles  = N / BN;
        int mblocks = (M / 16 + 7) / 8;
        return mblocks * ctiles;
    };

    // 1) v_proj = v @ W_in + b_in            [ROWS,256]x[256,256]
    gemm_bias_wmma_f32<<<gemm_grid(ROWS, DIMS), THREADS, 0, stream>>>(
        v, W_in, b_in, vproj, ROWS, DIMS);
    // 2) off = q @ W_off + b_off             [ROWS,256]x[256,256]
    gemm_bias_wmma_f32<<<gemm_grid(ROWS, DIMS), THREADS, 0, stream>>>(
        q, W_off, b_off, offw, ROWS, DIMS);
    // 3) attn logits = q @ W_attn + b_attn   [ROWS,256]x[256,128]
    gemm_bias_wmma_f32<<<gemm_grid(ROWS, HEADS * LEV * PTS), THREADS, 0, stream>>>(
        q, W_attn, b_attn, attnw, ROWS, HEADS * LEV * PTS);

    // 4) softmax + bilinear sampling: one wave per (n,q,head)
    const int total_waves = NB * QN * HEADS;     // 195,840
    msda_sample_kernel<<<total_waves / 8, THREADS, 0, stream>>>(
        vproj, offw, attnw, p, prew);

    // 5) out = sampled @ W_out + b_out       [ROWS,256]x[256,256]
    gemm_bias_wmma_f32<<<gemm_grid(ROWS, DIMS), THREADS, 0, stream>>>(
        prew, W_out, b_out, out, ROWS, DIMS);
}